// GroupNSTokenizer_2224793059932
// MI455X (gfx1250) — compile-verified
//
#include <hip/hip_runtime.h>
#include <hip/hip_bf16.h>

typedef __attribute__((ext_vector_type(2))) float v2f;
typedef __attribute__((ext_vector_type(4))) float v4f;
typedef __attribute__((ext_vector_type(8))) float v8f;

// Problem constants (from reference)
constexpr int kB       = 4096;
constexpr int kV       = 100001;
constexpr int kE       = 64;
constexpr int kD       = 512;
constexpr int kG       = 4;
constexpr int kIntDim  = 168;
constexpr int kKdim    = 256;   // 4 features * 64
constexpr float kLnEps = 1e-5f;

// Tiling: 32 batch rows per block -> 2 M-tiles share every B (W) fragment,
// halving L2 traffic on W versus a 16-row tile.
constexpr int TILE_B   = 32;
constexpr int A_STRIDE = 260;   // 256 + 4 pad (spreads LDS banks for ds_load_b64)
constexpr int H_STRIDE = 516;   // 512 + 4 pad

__global__ __launch_bounds__(256)
void tokenizer_fused(const int* __restrict__ int_feats,
                     const float* __restrict__ emb,      // (16, 100001, 64)
                     const float* __restrict__ W,        // (4, 256, 512)
                     const float* __restrict__ bias,     // (4, 512)
                     const float* __restrict__ gamma,    // (4, 512)
                     const float* __restrict__ beta,     // (4, 512)
                     float* __restrict__ out)            // (4096, 4, 512)
{
    __shared__ float Atile[TILE_B * A_STRIDE];   // gathered/pooled cat matrix, 32x256
    __shared__ float Htile[TILE_B * H_STRIDE];   // GEMM result h, 32x512

    const int tid  = threadIdx.x;
    const int g    = blockIdx.y;
    const int row0 = blockIdx.x * TILE_B;

    // ---------------- Phase 1: gather + masked mean-pool into LDS ----------------
    // 128 (row,feature) pairs, 2 threads each, 32 embedding dims per thread.
    {
        const int pair = tid >> 1;          // 0..127
        const int f    = pair >> 5;         // 0..3  feature within group
        const int row  = pair & 31;         // 0..31 batch row within tile
        const int half = tid & 1;           // dim half: [half*32, half*32+32)
        const int feat = g * 4 + f;
        // FEATURE_SPECS: even feat -> off=(feat/2)*21, len=1; odd -> off=(feat/2)*21+1, len=20
        const int off  = (feat >> 1) * 21 + (feat & 1);
        const int len  = (feat & 1) ? 20 : 1;

        const float* table = emb + (size_t)feat * ((size_t)kV * kE);
        const int*   ids   = int_feats + (size_t)(row0 + row) * kIntDim + off;

        float acc[32];
        #pragma unroll
        for (int d = 0; d < 32; ++d) acc[d] = 0.0f;
        int cnt = 0;

        for (int t = 0; t < len; ++t) {
            const int id = ids[t];
            if (id != 0) {                  // table row 0 is zeroed / masked out
                ++cnt;
                const float* src = table + (size_t)id * kE + half * 32;
                #pragma unroll
                for (int d4 = 0; d4 < 8; ++d4) {
                    v4f v = *(const v4f*)(src + d4 * 4);
                    acc[d4 * 4 + 0] += v.x;
                    acc[d4 * 4 + 1] += v.y;
                    acc[d4 * 4 + 2] += v.z;
                    acc[d4 * 4 + 3] += v.w;
                }
            }
        }
        const float scale = (len == 1) ? 1.0f : (1.0f / (float)(cnt > 0 ? cnt : 1));
        float* dst = &Atile[row * A_STRIDE + f * 64 + half * 32];
        #pragma unroll
        for (int d = 0; d < 32; ++d) dst[d] = acc[d] * scale;
    }
    __syncthreads();

    // ---------------- Phase 2: fp32 WMMA GEMM: (32x256) @ (256x512) ----------------
    const int wave  = tid >> 5;
    const int lane  = tid & 31;
    const int lhalf = lane >> 4;            // 0: K=k,k+1   1: K=k+2,k+3 (A); same split for B
    const int l16   = lane & 15;
    const float* Wg = W + (size_t)g * (kKdim * kD);

    #pragma unroll
    for (int i = 0; i < 4; ++i) {
        const int n_tile = wave + 8 * i;    // 32 N-tiles of 16 cols, 4 per wave
        const int ncol   = n_tile * 16 + l16;
        const float* A0  = &Atile[l16 * A_STRIDE];          // M-tile 0: rows 0-15
        const float* A1  = &Atile[(16 + l16) * A_STRIDE];   // M-tile 1: rows 16-31
        __builtin_prefetch((const void*)(Wg + ncol), 0, 3); // global_prefetch_b8, keep near

        v8f c0 = {0.f, 0.f, 0.f, 0.f, 0.f, 0.f, 0.f, 0.f};
        v8f c1 = {0.f, 0.f, 0.f, 0.f, 0.f, 0.f, 0.f, 0.f};
        #pragma unroll 4
        for (int k = 0; k < kKdim; k += 4) {
            // B fragment 4x16 f32 loaded once, reused by both M-tiles.
            const float* wp = Wg + (size_t)(k + lhalf * 2) * kD + ncol;
            v2f b;
            b.x = wp[0];
            b.y = wp[kD];
            // A fragments 16x4 f32: lanes 0-15 K=k..k+1, lanes 16-31 K=k+2..k+3
            v2f a0 = *(const v2f*)(A0 + k + lhalf * 2);
            v2f a1 = *(const v2f*)(A1 + k + lhalf * 2);
            c0 = __builtin_amdgcn_wmma_f32_16x16x4_f32(
                     false, a0, false, b, (short)0, c0, false, false);
            c1 = __builtin_amdgcn_wmma_f32_16x16x4_f32(
                     false, a1, false, b, (short)0, c1, false, false);
        }
        // C/D layout: VGPR r -> row (r + lhalf*8), col = l16
        #pragma unroll
        for (int r = 0; r < 8; ++r) {
            Htile[(r + lhalf * 8) * H_STRIDE + n_tile * 16 + l16]        = c0[r];
            Htile[(16 + r + lhalf * 8) * H_STRIDE + n_tile * 16 + l16]   = c1[r];
        }
    }
    __syncthreads();

    // ---------------- Phase 3: +bias, LayerNorm(512), SiLU, store ----------------
    const float* bg = bias  + (size_t)g * kD;
    const float* gg = gamma + (size_t)g * kD;
    const float* eg = beta  + (size_t)g * kD;

    #pragma unroll
    for (int rr = 0; rr < 4; ++rr) {        // 32 rows / 8 waves = 4 rows per wave
        const int row = wave * 4 + rr;
        float vals[16];
        float s = 0.f, s2 = 0.f;
        #pragma unroll
        for (int i = 0; i < 16; ++i) {
            const int d = lane + i * 32;
            const float v = Htile[row * H_STRIDE + d] + bg[d];
            vals[i] = v;
            s  += v;
            s2 += v * v;
        }
        // wave32 butterfly reduction
        #pragma unroll
        for (int m = 16; m >= 1; m >>= 1) {
            s  += __shfl_xor(s,  m, 32);
            s2 += __shfl_xor(s2, m, 32);
        }
        const float mu   = s  * (1.0f / (float)kD);
        const float var  = s2 * (1.0f / (float)kD) - mu * mu;
        const float rstd = rsqrtf(var + kLnEps);

        float* orow = out + ((size_t)(row0 + row) * kG + g) * kD;
        #pragma unroll
        for (int i = 0; i < 16; ++i) {
            const int d  = lane + i * 32;
            const float hn = (vals[i] - mu) * rstd * gg[d] + eg[d];
            orow[d] = hn / (1.0f + __expf(-hn));   // SiLU
        }
    }
}

extern "C" void kernel_launch(void* const* d_in, const int* in_sizes, int n_in,
                              void* d_out, int out_size, void* d_ws, size_t ws_size,
                              hipStream_t stream) {
    const int*   int_feats = (const int*)  d_in[0];
    const float* emb       = (const float*)d_in[1];
    const float* W         = (const float*)d_in[2];
    const float* bias      = (const float*)d_in[3];
    const float* gamma     = (const float*)d_in[4];
    const float* beta      = (const float*)d_in[5];
    float* out = (float*)d_out;

    dim3 grid(kB / TILE_B, kG, 1);
    tokenizer_fused<<<grid, 256, 0, stream>>>(int_feats, emb, W, bias, gamma, beta, out);
}